// SAGEConv_Decoder_85950885527593
// MI455X (gfx1250) — compile-verified
//
#include <hip/hip_runtime.h>
#include <math.h>

typedef __attribute__((ext_vector_type(2))) float v2f;
typedef __attribute__((ext_vector_type(8))) float v8f;

#define D 128
#define LDSP 132  // padded LDS row stride (floats) to avoid bank conflicts

// ---------------- utility kernels ----------------

__global__ __launch_bounds__(256) void zero_f32(float* __restrict__ p, int n) {
    int i = blockIdx.x * 256 + threadIdx.x;
    if (i < n) p[i] = 0.0f;
}

__global__ __launch_bounds__(256) void count_edges(const int* __restrict__ ei,
                                                   float* __restrict__ cnt, int E) {
    int e = blockIdx.x * 256 + threadIdx.x;
    if (e < E) unsafeAtomicAdd(&cnt[ei[E + e]], 1.0f);  // native global_atomic_add_f32
}

__global__ __launch_bounds__(256) void inv_count(const float* __restrict__ cnt,
                                                 float* __restrict__ inv, int n) {
    int i = blockIdx.x * 256 + threadIdx.x;
    if (i < n) inv[i] = 1.0f / fmaxf(cnt[i], 1.0f);
}

// One wave per edge: 32 lanes x float4 = 128 features. Hardware f32 atomics.
__global__ __launch_bounds__(256) void scatter_add(const float* __restrict__ x,
                                                   const int* __restrict__ ei,
                                                   float* __restrict__ agg, int E) {
    int e = blockIdx.x * 8 + (threadIdx.x >> 5);
    if (e >= E) return;
    int lane = threadIdx.x & 31;
    int src = ei[e];
    int dst = ei[E + e];
    const float4 v = *(const float4*)(x + (size_t)src * D + lane * 4);
    float* d = agg + (size_t)dst * D + lane * 4;
    unsafeAtomicAdd(d + 0, v.x);
    unsafeAtomicAdd(d + 1, v.y);
    unsafeAtomicAdd(d + 2, v.z);
    unsafeAtomicAdd(d + 3, v.w);
}

// ---------------- fused SAGE conv tile kernel ----------------
// out[16,128] = relu( l2norm_rows( mean_agg @ Wl + b + x @ Wr ) )
// 256 threads = 8 wave32; wave w owns output columns [16w, 16w+16).

__global__ __launch_bounds__(256) void sage_tile_kernel(
    const float* __restrict__ agg_raw, const float* __restrict__ inv_cnt,
    const float* __restrict__ xin, const float* __restrict__ Wl,
    const float* __restrict__ Wr, const float* __restrict__ bias,
    float* __restrict__ out, int N) {
    __shared__ float sA[16][LDSP];
    __shared__ float sX[16][LDSP];
    __shared__ float sO[16][LDSP];
    __shared__ float sRed[16][16];
    __shared__ float sScale[16];

    const int tid = threadIdx.x;
    const int node0 = blockIdx.x * 16;

    for (int i = tid; i < 16 * D; i += 256) {
        int r = i >> 7, c = i & (D - 1);
        int node = node0 + r;
        if (node >= N) node = N - 1;  // tail clamp (N%16==0 for this problem)
        sA[r][c] = agg_raw[(size_t)node * D + c] * inv_cnt[node];
        sX[r][c] = xin[(size_t)node * D + c];
    }
    __syncthreads();

    const int lane = tid & 31;
    const int j0 = (tid >> 5) * 16;  // wave's column tile
    const int m = lane & 15;
    const int half = lane >> 4;  // 0: K pair low, 1: K pair high
    const int j = j0 + m;

    v8f acc;
    {
        float bj = bias[j];
#pragma unroll
        for (int i = 0; i < 8; ++i) acc[i] = bj;
    }

#pragma unroll 4
    for (int k = 0; k < 32; ++k) {
        int kk = 4 * k + 2 * half;
        v2f aA = *(const v2f*)&sA[m][kk];  // A[m][kk], A[m][kk+1]
        v2f aX = *(const v2f*)&sX[m][kk];
        v2f bL, bR;
        bL[0] = Wl[(size_t)kk * D + j];
        bL[1] = Wl[(size_t)(kk + 1) * D + j];
        bR[0] = Wr[(size_t)kk * D + j];
        bR[1] = Wr[(size_t)(kk + 1) * D + j];
        acc = __builtin_amdgcn_wmma_f32_16x16x4_f32(false, aA, false, bL,
                                                    (short)0, acc, false, false);
        acc = __builtin_amdgcn_wmma_f32_16x16x4_f32(false, aX, false, bR,
                                                    (short)0, acc, false, false);
    }

// C layout: VGPR i -> row (i + 8*half), col j0+(lane&15)
#pragma unroll
    for (int i = 0; i < 8; ++i) sO[i + 8 * half][j] = acc[i];
    __syncthreads();

    // per-row sum of squares: 16 threads/row, 8 elems each
    {
        int r = tid >> 4, seg = tid & 15;
        float ssq = 0.0f;
#pragma unroll
        for (int c = 0; c < 8; ++c) {
            float v = sO[r][seg * 8 + c];
            ssq += v * v;
        }
        sRed[r][seg] = ssq;
    }
    __syncthreads();
    if (tid < 16) {
        float s = 0.0f;
#pragma unroll
        for (int i = 0; i < 16; ++i) s += sRed[tid][i];
        sScale[tid] = 1.0f / fmaxf(sqrtf(s), 1e-12f);
    }
    __syncthreads();

    for (int i = tid; i < 16 * D; i += 256) {
        int r = i >> 7, c = i & (D - 1);
        int node = node0 + r;
        if (node < N) {
            float v = sO[r][c] * sScale[r];
            out[(size_t)node * D + c] = fmaxf(v, 0.0f);
        }
    }
}

// ---------------- final FC tile kernel: out = h @ W + b ----------------

__global__ __launch_bounds__(256) void fc_tile_kernel(const float* __restrict__ h,
                                                      const float* __restrict__ W,
                                                      const float* __restrict__ bias,
                                                      float* __restrict__ out, int N) {
    __shared__ float sA[16][LDSP];
    const int tid = threadIdx.x;
    const int node0 = blockIdx.x * 16;

    for (int i = tid; i < 16 * D; i += 256) {
        int r = i >> 7, c = i & (D - 1);
        int node = node0 + r;
        if (node >= N) node = N - 1;
        sA[r][c] = h[(size_t)node * D + c];
    }
    __syncthreads();

    const int lane = tid & 31;
    const int j0 = (tid >> 5) * 16;
    const int m = lane & 15;
    const int half = lane >> 4;
    const int j = j0 + m;

    v8f acc;
    {
        float bj = bias[j];
#pragma unroll
        for (int i = 0; i < 8; ++i) acc[i] = bj;
    }

#pragma unroll 4
    for (int k = 0; k < 32; ++k) {
        int kk = 4 * k + 2 * half;
        v2f a = *(const v2f*)&sA[m][kk];
        v2f b;
        b[0] = W[(size_t)kk * D + j];
        b[1] = W[(size_t)(kk + 1) * D + j];
        acc = __builtin_amdgcn_wmma_f32_16x16x4_f32(false, a, false, b, (short)0,
                                                    acc, false, false);
    }

#pragma unroll
    for (int i = 0; i < 8; ++i) {
        int node = node0 + i + 8 * half;
        if (node < N) out[(size_t)node * D + j] = acc[i];
    }
}

// ---------------- host launch ----------------

extern "C" void kernel_launch(void* const* d_in, const int* in_sizes, int n_in,
                              void* d_out, int out_size, void* d_ws, size_t ws_size,
                              hipStream_t stream) {
    const float* x = (const float*)d_in[0];
    const int* ei = (const int*)d_in[1];
    const float* W1l = (const float*)d_in[2];
    const float* b1 = (const float*)d_in[3];
    const float* W1r = (const float*)d_in[4];
    const float* W2l = (const float*)d_in[5];
    const float* b2 = (const float*)d_in[6];
    const float* W2r = (const float*)d_in[7];
    const float* Wf = (const float*)d_in[8];
    const float* bf = (const float*)d_in[9];
    float* outp = (float*)d_out;

    const int N = in_sizes[0] / D;
    const int E = in_sizes[1] / 2;
    const int ND = N * D;

    float* ws = (float*)d_ws;
    float* cnt = ws;
    float* inv = ws + N;
    float* agg = ws + 2 * (size_t)N;
    float* h1 = agg + (size_t)N * D;
    float* h2 = h1 + (size_t)N * D;

    const int nTiles = (N + 15) / 16;

    // degrees + reciprocal mean factors
    zero_f32<<<(N + 255) / 256, 256, 0, stream>>>(cnt, N);
    count_edges<<<(E + 255) / 256, 256, 0, stream>>>(ei, cnt, E);
    inv_count<<<(N + 255) / 256, 256, 0, stream>>>(cnt, inv, N);

    // layer 1
    zero_f32<<<(ND + 255) / 256, 256, 0, stream>>>(agg, ND);
    scatter_add<<<(E + 7) / 8, 256, 0, stream>>>(x, ei, agg, E);
    sage_tile_kernel<<<nTiles, 256, 0, stream>>>(agg, inv, x, W1l, W1r, b1, h1, N);

    // layer 2
    zero_f32<<<(ND + 255) / 256, 256, 0, stream>>>(agg, ND);
    scatter_add<<<(E + 7) / 8, 256, 0, stream>>>(h1, ei, agg, E);
    sage_tile_kernel<<<nTiles, 256, 0, stream>>>(agg, inv, h1, W2l, W2r, b2, h2, N);

    // final fc
    fc_tile_kernel<<<nTiles, 256, 0, stream>>>(h2, Wf, bf, outp, N);
}